// MGCN_24747601560207
// MI455X (gfx1250) — compile-verified
//
#include <hip/hip_runtime.h>

#define NUM_USER   100000
#define NUM_ITEM   50000
#define NTOT       (NUM_USER + NUM_ITEM)
#define DIM_FEAT   512
#define DIM_LATENT 64

typedef __attribute__((ext_vector_type(2))) float v2f;
typedef __attribute__((ext_vector_type(8))) float v8f;

// ---------------------------------------------------------------------------
// Kernel 1: temp_features = features @ W + b  via V_WMMA_F32_16X16X4_F32.
// One wave computes one 16x16 output tile; block = 4 waves = the 4 N-tiles
// (N = 64) of one 16-row M tile. Grid = 50000/16 = 3125 blocks.
// A layout (16x4 f32): lanes 0-15 -> K = {k0, k0+1}, lanes 16-31 -> {k0+2, k0+3}
// (contiguous float2 from the row-major feature row). B mirrors per-column.
// C/D: v8f, element j -> row (j + 8*half), col nbase + (lane&15).
// ---------------------------------------------------------------------------
__global__ __launch_bounds__(128) void mgcn_gemm_wmma(
    const float* __restrict__ F, const float* __restrict__ W,
    const float* __restrict__ bias, float* __restrict__ emb)
{
    const int wave  = threadIdx.x >> 5;     // 0..3 -> N tile
    const int lane  = threadIdx.x & 31;
    const int half  = lane >> 4;            // 0 or 1
    const int mrow  = lane & 15;
    const int mtile = blockIdx.x;           // 0..3124
    const int m     = mtile * 16 + mrow;
    const int ncol  = wave * 16 + (lane & 15);

    const float* frow = F + (size_t)m * DIM_FEAT;

    v8f c = {};
    for (int k0 = 0; k0 < DIM_FEAT; k0 += 4) {
        const int k = k0 + 2 * half;
        // A: aligned float2 (k0 % 4 == 0, 2*half even -> 8B aligned)
        v2f a = *(const v2f*)(frow + k);
        // B: W is row-major (K x N); two strided scalar loads per lane
        v2f bm;
        bm.x = W[(size_t)k * DIM_LATENT + ncol];
        bm.y = W[(size_t)(k + 1) * DIM_LATENT + ncol];
        c = __builtin_amdgcn_wmma_f32_16x16x4_f32(
                /*neg_a=*/false, a, /*neg_b=*/false, bm,
                /*c_mod=*/(short)0, c, /*reuse_a=*/false, /*reuse_b=*/false);
    }

    const float bv = bias[ncol];
    float* dst = emb + (size_t)(NUM_USER + mtile * 16) * DIM_LATENT;
#pragma unroll
    for (int j = 0; j < 8; ++j) {
        const int mr = j + 8 * half;
        dst[(size_t)mr * DIM_LATENT + ncol] = c[j] + bv;
    }
}

// ---------------------------------------------------------------------------
// Kernel 2: L2-normalize each 64-wide row (users from user_id_preference,
// items already written raw by the GEMM), write normalized row into the
// embedding buffer AND initialize embs_sum (= d_out). One wave per row:
// lane holds a float2, 5-step wave32 xor-shuffle reduction for the norm.
// ---------------------------------------------------------------------------
__global__ __launch_bounds__(256) void mgcn_normalize_init(
    const float* __restrict__ upref, float* __restrict__ emb,
    float* __restrict__ sum)
{
    const int row  = (int)((blockIdx.x * blockDim.x + threadIdx.x) >> 5);
    const int lane = threadIdx.x & 31;
    if (row >= NTOT) return;

    const float* src = (row < NUM_USER)
                           ? (upref + (size_t)row * DIM_LATENT)
                           : (emb + (size_t)row * DIM_LATENT);
    v2f x = *(const v2f*)(src + lane * 2);

    float s = x.x * x.x + x.y * x.y;
#pragma unroll
    for (int off = 16; off; off >>= 1) s += __shfl_xor(s, off, 32);

    const float inv = 1.0f / fmaxf(sqrtf(s), 1e-12f);
    v2f y; y.x = x.x * inv; y.y = x.y * inv;

    *(v2f*)(emb + (size_t)row * DIM_LATENT + lane * 2) = y;
    *(v2f*)(sum + (size_t)row * DIM_LATENT + lane * 2) = y;
}

// ---------------------------------------------------------------------------
// Kernel 3: one SpMM layer: dst[rows[e]] += src[cols[e]] * vals[e].
// One wave processes 32 edges: edge indices loaded coalesced (one per lane),
// then broadcast with wave32 shuffles. Per edge: coalesced float2 gather
// (32 lanes x 2 = 64 dims, table is L2-resident) + 2 f32 global atomics/lane.
// ---------------------------------------------------------------------------
__global__ __launch_bounds__(256) void mgcn_spmm_scatter(
    const float* __restrict__ src, const int* __restrict__ rows,
    const int* __restrict__ cols, const float* __restrict__ vals,
    float* __restrict__ dst, int nnz)
{
    const int wave = (int)((blockIdx.x * blockDim.x + threadIdx.x) >> 5);
    const int lane = threadIdx.x & 31;
    const int base = wave * 32;
    if (base >= nnz) return;

    const int myE = base + lane;
    int   r32 = 0, c32 = 0;
    float v32 = 0.0f;
    if (myE < nnz) { r32 = rows[myE]; c32 = cols[myE]; v32 = vals[myE]; }

    const int cnt = (nnz - base < 32) ? (nnz - base) : 32;
    for (int i = 0; i < cnt; ++i) {
        const int   r = __shfl(r32, i, 32);
        const int   c = __shfl(c32, i, 32);
        const float v = __shfl(v32, i, 32);

        v2f x = *(const v2f*)(src + (size_t)c * DIM_LATENT + lane * 2);
        float* d = dst + (size_t)r * DIM_LATENT + lane * 2;
        atomicAdd(d,     x.x * v);
        atomicAdd(d + 1, x.y * v);
    }
}

// ---------------------------------------------------------------------------
// Kernel 4: embs_sum = (embs_sum + layer_result) * scale (scale = 0.25 folds
// the final /(N_LAYERS+1) into the last layer's accumulation). float4 I/O.
// ---------------------------------------------------------------------------
__global__ __launch_bounds__(256) void mgcn_accumulate4(
    float4* __restrict__ sum, const float4* __restrict__ add,
    float scale, int n4)
{
    const int i = blockIdx.x * blockDim.x + threadIdx.x;
    if (i >= n4) return;
    float4 s = sum[i];
    const float4 a = add[i];
    s.x = (s.x + a.x) * scale;
    s.y = (s.y + a.y) * scale;
    s.z = (s.z + a.z) * scale;
    s.w = (s.w + a.w) * scale;
    sum[i] = s;
}

// ---------------------------------------------------------------------------
// Driver: d_out doubles as embs_sum (users||items flattened == embs_sum/4).
// d_ws holds two ping-pong embedding buffers (2 x 38.4 MB).
// ---------------------------------------------------------------------------
extern "C" void kernel_launch(void* const* d_in, const int* in_sizes, int n_in,
                              void* d_out, int out_size, void* d_ws, size_t ws_size,
                              hipStream_t stream)
{
    const float* features = (const float*)d_in[0];
    const float* upref    = (const float*)d_in[1];
    const float* W        = (const float*)d_in[2];
    const float* bias     = (const float*)d_in[3];
    const int*   rows     = (const int*)d_in[4];
    const int*   cols     = (const int*)d_in[5];
    const float* vals     = (const float*)d_in[6];
    float*       out      = (float*)d_out;
    const int    nnz      = in_sizes[4];

    const size_t embBytes = (size_t)NTOT * DIM_LATENT * sizeof(float);
    float* bufA = (float*)d_ws;
    float* bufB = (float*)((char*)d_ws + embBytes);

    // 1) GEMM: raw item embeddings into bufA rows [NUM_USER, NTOT)
    mgcn_gemm_wmma<<<NUM_ITEM / 16, 128, 0, stream>>>(features, W, bias, bufA);

    // 2) Normalize all rows into bufA; initialize embs_sum (= d_out)
    {
        const long long threads = (long long)NTOT * 32;
        const int blocks = (int)((threads + 255) / 256);
        mgcn_normalize_init<<<blocks, 256, 0, stream>>>(upref, bufA, out);
    }

    // 3) Three propagation layers with ping-pong buffers
    float* cur = bufA;
    float* nxt = bufB;
    const int scatterWaves  = (nnz + 31) / 32;
    const int scatterBlocks = (int)(((long long)scatterWaves * 32 + 255) / 256);
    const int n4            = NTOT * DIM_LATENT / 4;
    const int accBlocks     = (n4 + 255) / 256;

    for (int layer = 0; layer < 3; ++layer) {
        hipMemsetAsync(nxt, 0, embBytes, stream);
        mgcn_spmm_scatter<<<scatterBlocks, 256, 0, stream>>>(cur, rows, cols,
                                                             vals, nxt, nnz);
        const float scale = (layer == 2) ? 0.25f : 1.0f;
        mgcn_accumulate4<<<accBlocks, 256, 0, stream>>>((float4*)out,
                                                        (const float4*)nxt,
                                                        scale, n4);
        float* t = cur; cur = nxt; nxt = t;
    }
}